// SEP_U_synthetic_84988812853302
// MI455X (gfx1250) — compile-verified
//
#include <hip/hip_runtime.h>

#define NN 100000
#define CC 25000
#define EE 1000000
#define DD 64
#define BN_EPS 1e-5f

typedef float v2f __attribute__((ext_vector_type(2)));
typedef float v8f __attribute__((ext_vector_type(8)));

// ---------------------------------------------------------------- utilities
__global__ void fill_kernel(float* __restrict__ p, int n, float v) {
    int i = blockIdx.x * blockDim.x + threadIdx.x;
    if (i < n) p[i] = v;
}

__global__ void deg_kernel(const int* __restrict__ col, float* __restrict__ deg, int e) {
    int i = blockIdx.x * blockDim.x + threadIdx.x;
    if (i < e) atomicAdd(deg + col[i], 1.0f);
}

__global__ void rsqrt_kernel(float* __restrict__ p, int n) {
    int i = blockIdx.x * blockDim.x + threadIdx.x;
    if (i < n) p[i] = rsqrtf(p[i]);
}

__global__ void relu_kernel(float* __restrict__ p, int n) {
    int i = blockIdx.x * blockDim.x + threadIdx.x;
    if (i < n) p[i] = fmaxf(p[i], 0.0f);
}

// ---------------------------------------------------------------- WMMA GEMM
// out[r][:] = (rowidx ? x[rowidx[r]] : x[r]) @ W   (+ bias, ReLU if bias!=null)
// W is DDxDD row-major. 256 threads = 8 waves; each wave produces a 16x64 tile
// via 4 accumulators of V_WMMA_F32_16X16X4_F32, K=64 in 16 steps.
// Tail rows: A source row is CLAMPED (not zeroed) -> unconditional loads, no
// EXEC churn in the inner loop; garbage rows are discarded by the store guard
// (A row M only contributes to C row M).
__global__ __launch_bounds__(256)
void gemm64_wmma_kernel(const float* __restrict__ x,
                        const float* __restrict__ W,
                        const float* __restrict__ bias,
                        const int* __restrict__ rowidx,
                        float* __restrict__ out,
                        int nrows)
{
    __shared__ float sW[DD * DD];   // 16 KB
    const int t = threadIdx.x;
#pragma unroll
    for (int i = 0; i < (DD * DD) / 256; ++i)
        sW[t + i * 256] = W[t + i * 256];
    __syncthreads();

    const int wave = t >> 5;
    const int lane = t & 31;
    const int hh   = lane >> 4;     // half-wave: 0 or 1
    const int l16  = lane & 15;
    const int rowBase = (blockIdx.x * 8 + wave) * 16;

    // A-fragment source row for this lane (clamped for the tail tile)
    int arow = rowBase + l16;
    if (arow > nrows - 1) arow = nrows - 1;
    const int asrc = rowidx ? rowidx[arow] : arow;
    const float* aptr = x + (size_t)asrc * DD;

    v8f acc0 = {}, acc1 = {}, acc2 = {}, acc3 = {};

#pragma unroll
    for (int k0 = 0; k0 < DD; k0 += 4) {
        // A 16x4 f32: VGPR0/1 = K={kk,kk+1}; lanes 0-15 -> kk=k0, lanes 16-31 -> kk=k0+2
        const int kk = k0 + hh * 2;
        v2f a = *(const v2f*)(aptr + kk);

        // B 4x16 f32: row-striped across lanes; VGPR0 = K=kk row, VGPR1 = K=kk+1 row
        v2f b0, b1, b2, b3;
        b0.x = sW[kk * DD + l16 +  0];  b0.y = sW[(kk + 1) * DD + l16 +  0];
        b1.x = sW[kk * DD + l16 + 16];  b1.y = sW[(kk + 1) * DD + l16 + 16];
        b2.x = sW[kk * DD + l16 + 32];  b2.y = sW[(kk + 1) * DD + l16 + 32];
        b3.x = sW[kk * DD + l16 + 48];  b3.y = sW[(kk + 1) * DD + l16 + 48];

        acc0 = __builtin_amdgcn_wmma_f32_16x16x4_f32(false, a, false, b0, (short)0, acc0, false, false);
        acc1 = __builtin_amdgcn_wmma_f32_16x16x4_f32(false, a, false, b1, (short)0, acc1, false, false);
        acc2 = __builtin_amdgcn_wmma_f32_16x16x4_f32(false, a, false, b2, (short)0, acc2, false, false);
        acc3 = __builtin_amdgcn_wmma_f32_16x16x4_f32(false, a, false, b3, (short)0, acc3, false, false);
    }

    const bool dorelu = (bias != nullptr);
    float bb0 = 0.f, bb1 = 0.f, bb2 = 0.f, bb3 = 0.f;
    if (dorelu) {
        bb0 = bias[l16];      bb1 = bias[l16 + 16];
        bb2 = bias[l16 + 32]; bb3 = bias[l16 + 48];
    }

    // C/D 16x16 f32: VGPR v -> M = v (lanes 0-15) or v+8 (lanes 16-31), N = l16
#pragma unroll
    for (int v = 0; v < 8; ++v) {
        const int r = rowBase + hh * 8 + v;
        if (r < nrows) {
            float o0 = acc0[v], o1 = acc1[v], o2 = acc2[v], o3 = acc3[v];
            if (dorelu) {
                o0 = fmaxf(o0 + bb0, 0.f); o1 = fmaxf(o1 + bb1, 0.f);
                o2 = fmaxf(o2 + bb2, 0.f); o3 = fmaxf(o3 + bb3, 0.f);
            }
            float* op = out + (size_t)r * DD + l16;
            op[0] = o0; op[16] = o1; op[32] = o2; op[48] = o3;
        }
    }
}

// ---------------------------------------------------------------- GCN conv pieces
// out[n][d] = bias[d] + h[n][d] * dinv[n]^2   (self-loop + bias)
__global__ void conv_init_kernel(const float* __restrict__ h,
                                 const float* __restrict__ dinv,
                                 const float* __restrict__ bias,
                                 float* __restrict__ out, int nnodes)
{
    int gid = blockIdx.x * blockDim.x + threadIdx.x;
    if (gid >= nnodes * DD) return;
    int node = gid >> 6, d = gid & 63;
    float di = dinv[node];
    out[gid] = bias[d] + h[gid] * di * di;
}

// per edge e: out[col[e]][:] += h[row[e]][:] * dinv[row]*dinv[col]
// thread handles 4 consecutive dims of one edge (16 threads/edge)
__global__ void conv_scatter_kernel(const float* __restrict__ h,
                                    const int* __restrict__ rows,
                                    const int* __restrict__ cols,
                                    const float* __restrict__ dinv,
                                    float* __restrict__ out, int nedges)
{
    int gid = blockIdx.x * blockDim.x + threadIdx.x;
    int e = gid >> 4;
    if (e >= nedges) return;
    int q = gid & 15;
    int r = rows[e], c = cols[e];
    float w = dinv[r] * dinv[c];
    float4 hv = *(const float4*)(h + (size_t)r * DD + q * 4);
    float* op = out + (size_t)c * DD + q * 4;
    atomicAdd(op + 0, hv.x * w);
    atomicAdd(op + 1, hv.y * w);
    atomicAdd(op + 2, hv.z * w);
    atomicAdd(op + 3, hv.w * w);
}

// ---------------------------------------------------------------- pooling pieces
__global__ void pool_scatter_kernel(const float* __restrict__ x,
                                    const int* __restrict__ clus,
                                    float* __restrict__ agg, int nnodes)
{
    int gid = blockIdx.x * blockDim.x + threadIdx.x;
    if (gid >= nnodes * DD) return;
    int node = gid >> 6, d = gid & 63;
    atomicAdd(agg + (size_t)clus[node] * DD + d, x[gid]);
}

// per-column sum & sumsq over rows, coalesced; partials reduced in LDS then
// atomically accumulated into gsum[64]/gssq[64]
__global__ __launch_bounds__(256)
void bn_stats_kernel(const float* __restrict__ h,
                     float* __restrict__ gsum, float* __restrict__ gssq, int rowsTotal)
{
    __shared__ float ss[256];
    __shared__ float sq[256];
    int t = threadIdx.x;
    int col = t & 63;
    int rph = t >> 6;
    float s = 0.f, q = 0.f;
    for (int r = blockIdx.x * 4 + rph; r < rowsTotal; r += gridDim.x * 4) {
        float v = h[(size_t)r * DD + col];
        s += v; q += v * v;
    }
    ss[t] = s; sq[t] = q;
    __syncthreads();
    if (t < 128) { ss[t] += ss[t + 128]; sq[t] += sq[t + 128]; }
    __syncthreads();
    if (t < 64) {
        atomicAdd(gsum + t, ss[t] + ss[t + 64]);
        atomicAdd(gssq + t, sq[t] + sq[t + 64]);
    }
}

// out = relu( gamma*(h-mean)*rsqrt(var+eps) + beta ), stats finalized inline
__global__ void bn_apply_kernel(const float* __restrict__ h,
                                const float* __restrict__ gsum,
                                const float* __restrict__ gssq,
                                const float* __restrict__ gamma,
                                const float* __restrict__ beta,
                                float* __restrict__ out, int rowsTotal)
{
    int gid = blockIdx.x * blockDim.x + threadIdx.x;
    if (gid >= rowsTotal * DD) return;
    int d = gid & 63;
    float invR = 1.0f / (float)rowsTotal;
    float mean = gsum[d] * invR;
    float var  = gssq[d] * invR - mean * mean;
    float rstd = rsqrtf(var + BN_EPS);
    float v = gamma[d] * (h[gid] - mean) * rstd + beta[d];
    out[gid] = fmaxf(v, 0.0f);
}

// ---------------------------------------------------------------- launcher
extern "C" void kernel_launch(void* const* d_in, const int* in_sizes, int n_in,
                              void* d_out, int out_size, void* d_ws, size_t ws_size,
                              hipStream_t stream)
{
    (void)in_sizes; (void)n_in; (void)out_size; (void)ws_size;

    const float* x0    = (const float*)d_in[0];
    const int*   ei    = (const int*)d_in[1];
    const int*   sclus = (const int*)d_in[2];
    const float* convw = (const float*)d_in[3];
    const float* convb = (const float*)d_in[4];
    const float* poolw = (const float*)d_in[5];
    const float* poolb = (const float*)d_in[6];
    const float* gamma = (const float*)d_in[7];
    const float* beta  = (const float*)d_in[8];
    const int* rows = ei;        // edge_index[0] = source
    const int* cols = ei + EE;   // edge_index[1] = target

    char* ws = (char*)d_ws;
    float* dinv = (float*)(ws);                 // N floats
    float* hbuf = (float*)(ws + 400384);        // N*64 floats
    float* abuf = (float*)(ws + 26000384);      // C*64 floats
    float* bns  = (float*)(ws + 32400384);      // 64 sums
    float* bnq  = bns + 64;                     // 64 sumsq
    float* X    = (float*)d_out;                // ping buffer == output

    const int TB = 256;
    const int gN   = (NN + TB - 1) / TB;
    const int gE   = (EE + TB - 1) / TB;
    const int gND  = (NN * DD + TB - 1) / TB;
    const int gCD  = (CC * DD + TB - 1) / TB;
    const int gSc  = (EE * 16 + TB - 1) / TB;
    const int gGmN = (NN + 127) / 128;          // gemm: 128 rows per block
    const int gGmC = (CC + 127) / 128;

    // ---- degree -> dinv (self-loops via init 1.0)
    fill_kernel<<<gN, TB, 0, stream>>>(dinv, NN, 1.0f);
    deg_kernel<<<gE, TB, 0, stream>>>(cols, dinv, EE);
    rsqrt_kernel<<<gN, TB, 0, stream>>>(dinv, NN);

    auto conv = [&](const float* xin, int wi, bool do_relu) {
        gemm64_wmma_kernel<<<gGmN, TB, 0, stream>>>(xin, convw + wi * DD * DD,
                                                    nullptr, nullptr, hbuf, NN);
        conv_init_kernel<<<gND, TB, 0, stream>>>(hbuf, dinv, convb + wi * DD, X, NN);
        conv_scatter_kernel<<<gSc, TB, 0, stream>>>(hbuf, rows, cols, dinv, X, EE);
        if (do_relu) relu_kernel<<<gND, TB, 0, stream>>>(X, NN * DD);
    };

    // ---- blocks 0..1
    conv(x0, 0, true);
    conv(X,  1, true);

    // ---- pool down: nodes -> clusters
    fill_kernel<<<gCD, TB, 0, stream>>>(abuf, CC * DD, 0.0f);
    pool_scatter_kernel<<<gND, TB, 0, stream>>>(X, sclus, abuf, NN);
    gemm64_wmma_kernel<<<gGmC, TB, 0, stream>>>(abuf, poolw, poolb, nullptr, hbuf, CC);
    fill_kernel<<<1, 128, 0, stream>>>(bns, 128, 0.0f);
    bn_stats_kernel<<<512, TB, 0, stream>>>(hbuf, bns, bnq, CC);
    bn_apply_kernel<<<gCD, TB, 0, stream>>>(hbuf, bns, bnq, gamma, beta, abuf, CC);

    // ---- pool up: clusters -> nodes (gather fused into GEMM A-load)
    gemm64_wmma_kernel<<<gGmN, TB, 0, stream>>>(abuf, poolw + DD * DD, poolb + DD,
                                                sclus, hbuf, NN);
    fill_kernel<<<1, 128, 0, stream>>>(bns, 128, 0.0f);
    bn_stats_kernel<<<512, TB, 0, stream>>>(hbuf, bns, bnq, NN);
    bn_apply_kernel<<<gND, TB, 0, stream>>>(hbuf, bns, bnq, gamma + DD, beta + DD, X, NN);

    // ---- blocks 2..3 and final conv (no relu)
    conv(X, 2, true);
    conv(X, 3, true);
    conv(X, 4, false);
}